// EdgePrompt_plus_20392504721413
// MI455X (gfx1250) — compile-verified
//
#include <hip/hip_runtime.h>
#include <hip/hip_bf16.h>

typedef __attribute__((ext_vector_type(2))) float v2f;
typedef __attribute__((ext_vector_type(8))) float v8f;

#define N_NODES   50000
#define N_EDGES   800000
#define IN_DIM    128
#define NUM_BASES 5
#define D2        256
#define NCOL      16              // 12 useful columns padded to 16
#define NTILES    (N_NODES / 16)  // 3125 exact

// ---------------------------------------------------------------------------
// Kernel 1: build the packed projection matrix Wt[128][16] (K-major rows) and
// bdot[k] = dot(bases[k], scorer_w).  Columns of Wt:
//   j=0..4  : attn_w[j][0:128]      (src half)
//   j=5     : scorer_w[0:128]       (src half)
//   j=6..10 : attn_w[j-6][128:256]  (dst half)
//   j=11    : scorer_w[128:256]     (dst half)
//   j=12..15: zero padding
// ---------------------------------------------------------------------------
__global__ void ep_prep_kernel(const float* __restrict__ attn_w,
                               const float* __restrict__ scorer_w,
                               const float* __restrict__ bases,
                               float* __restrict__ Wt,
                               float* __restrict__ bdot) {
    int t = threadIdx.x;
    for (int i = t; i < IN_DIM * NCOL; i += blockDim.x) {
        int k = i >> 4;
        int j = i & 15;
        float v = 0.0f;
        if (j < NUM_BASES)                 v = attn_w[j * D2 + k];
        else if (j == 5)                   v = scorer_w[k];
        else if (j >= 6 && j < 6 + NUM_BASES) v = attn_w[(j - 6) * D2 + IN_DIM + k];
        else if (j == 11)                  v = scorer_w[IN_DIM + k];
        Wt[i] = v;
    }
    if (t < NUM_BASES) {
        float s = 0.0f;
        for (int d = 0; d < D2; ++d) s += bases[t * D2 + d] * scorer_w[d];
        bdot[t] = s;
    }
}

// ---------------------------------------------------------------------------
// Kernel 2: node projection GEMM  P[50000][16] = x[50000][128] * Wt[128][16]
// via V_WMMA_F32_16X16X4_F32 (full-precision f32 WMMA, 32 K-steps of 4).
// One 16-node tile per wave, 4 waves (128 threads) per block sharing an LDS
// copy of Wt.  EXEC is all-ones inside the WMMA region (tile guard is
// wave-uniform and taken before any WMMA).
//
// f32 16x16x4 operand layouts (ISA 7.12.2):
//   A 16x4:  lanes 0-15 -> M=lane, VGPR{0,1}=K{0,1}; lanes 16-31 -> K{2,3}
//   B 4x16:  lanes 0-15 -> N=lane, VGPR{0,1}=K{0,1}; lanes 16-31 -> K{2,3}
//   C/D 16x16: VGPR v -> M = v + 8*(lane>=16), N = lane&15
// ---------------------------------------------------------------------------
__global__ void __launch_bounds__(128)
ep_node_proj_kernel(const float* __restrict__ x,
                    const float* __restrict__ Wt,
                    float* __restrict__ P) {
    __shared__ float ldsW[IN_DIM * NCOL];   // 8 KB
    for (int i = threadIdx.x; i < IN_DIM * NCOL; i += 128) ldsW[i] = Wt[i];
    __syncthreads();

    const int wave = threadIdx.x >> 5;
    const int tile = blockIdx.x * 4 + wave;
    if (tile >= NTILES) return;             // wave-uniform exit

    const int lane = threadIdx.x & 31;
    const int half = lane >> 4;             // 0: K pair {0,1}, 1: K pair {2,3}
    const int l15  = lane & 15;

    const float* xrow  = x + (size_t)(tile * 16 + l15) * IN_DIM + 2 * half;
    const float* wbase = ldsW + (2 * half) * NCOL + l15;

    v8f c = {0.f, 0.f, 0.f, 0.f, 0.f, 0.f, 0.f, 0.f};
#pragma unroll
    for (int k0 = 0; k0 < IN_DIM; k0 += 4) {
        v2f a, b;
        // contiguous, 8B-aligned pair from this lane's node row
        float2 av = *(const float2*)(xrow + k0);
        a.x = av.x;
        a.y = av.y;
        // two K-rows of the LDS weight tile (stride NCOL floats)
        b.x = wbase[(k0 + 0) * NCOL];
        b.y = wbase[(k0 + 1) * NCOL];
        c = __builtin_amdgcn_wmma_f32_16x16x4_f32(
                /*neg_a=*/false, a, /*neg_b=*/false, b,
                /*c_mod=*/(short)0, c, /*reuse_a=*/false, /*reuse_b=*/false);
    }

    // D layout: VGPR v -> row M = v + 8*half, col N = l15
    float* prow = P + (size_t)(tile * 16 + 8 * half) * NCOL + l15;
#pragma unroll
    for (int v = 0; v < 8; ++v) prow[v * NCOL] = c[v];
}

// ---------------------------------------------------------------------------
// Kernel 3: per-edge pass.  Reads 6 floats from P[src] (cols 0-5) and 6 from
// P[dst] (cols 6-11); softmax over 5 attention logits; scorer value folded in
// via precomputed bdot[]; sigmoid; scale by edge_weight.
// ---------------------------------------------------------------------------
__global__ void __launch_bounds__(256)
ep_edge_kernel(const long long* __restrict__ edge_index,
               const float* __restrict__ edge_weight,
               const float* __restrict__ attn_b,
               const float* __restrict__ scorer_b,
               const float* __restrict__ P,
               const float* __restrict__ bdot,
               float* __restrict__ out) {
    int e = blockIdx.x * 256 + threadIdx.x;
    if (e >= N_EDGES) return;

    const int src = (int)edge_index[e];
    const int dst = (int)edge_index[(size_t)N_EDGES + e];

    const float* Ps = P + (size_t)src * NCOL;       // cols 0..5
    const float* Pd = P + (size_t)dst * NCOL + 6;   // cols 6..11

    float ps[6], pd[6];
    {
        float4 q = *(const float4*)Ps;              // 16B aligned
        float2 r = *(const float2*)(Ps + 4);
        ps[0] = q.x; ps[1] = q.y; ps[2] = q.z; ps[3] = q.w;
        ps[4] = r.x; ps[5] = r.y;
        float2 a0 = *(const float2*)(Pd + 0);       // 8B aligned (offset 24B)
        float2 a1 = *(const float2*)(Pd + 2);
        float2 a2 = *(const float2*)(Pd + 4);
        pd[0] = a0.x; pd[1] = a0.y; pd[2] = a1.x;
        pd[3] = a1.y; pd[4] = a2.x; pd[5] = a2.y;
    }

    float lg[NUM_BASES];
    float m = -3.0e38f;
#pragma unroll
    for (int k = 0; k < NUM_BASES; ++k) {
        lg[k] = ps[k] + pd[k] + attn_b[k];
        m = fmaxf(m, lg[k]);
    }
    float s0 = ps[5] + pd[5] + scorer_b[0];

    float sum = 0.0f, acc = 0.0f;
#pragma unroll
    for (int k = 0; k < NUM_BASES; ++k) {
        float w = expf(lg[k] - m);
        sum += w;
        acc += w * bdot[k];
    }
    float total = s0 + acc / sum;
    float sig   = 1.0f / (1.0f + expf(-total));
    out[e] = edge_weight[e] * sig;
}

// ---------------------------------------------------------------------------
extern "C" void kernel_launch(void* const* d_in, const int* in_sizes, int n_in,
                              void* d_out, int out_size, void* d_ws, size_t ws_size,
                              hipStream_t stream) {
    const float*     x           = (const float*)d_in[0];
    const long long* edge_index  = (const long long*)d_in[1];
    const float*     edge_weight = (const float*)d_in[2];
    const float*     bases       = (const float*)d_in[3];
    const float*     attn_w      = (const float*)d_in[4];
    const float*     attn_b      = (const float*)d_in[5];
    const float*     scorer_w    = (const float*)d_in[6];
    const float*     scorer_b    = (const float*)d_in[7];

    // workspace layout (floats): P[50000*16] | Wt[128*16] | bdot[8]
    float* P    = (float*)d_ws;
    float* Wt   = P + (size_t)N_NODES * NCOL;
    float* bdot = Wt + IN_DIM * NCOL;

    ep_prep_kernel<<<1, 256, 0, stream>>>(attn_w, scorer_w, bases, Wt, bdot);
    ep_node_proj_kernel<<<(NTILES + 3) / 4, 128, 0, stream>>>(x, Wt, P);
    ep_edge_kernel<<<(N_EDGES + 255) / 256, 256, 0, stream>>>(
        edge_index, edge_weight, attn_b, scorer_b, P, bdot, (float*)d_out);
}